// LevermoreEquilibrium_18760417149375
// MI455X (gfx1250) — compile-verified
//
#include <hip/hip_runtime.h>
#include <stdint.h>

#ifndef __has_builtin
#define __has_builtin(x) 0
#endif

#define CV_F 1.0f
#define RTOL_F 1e-8f
#define ATOL_F 1e-8f
#define MAX_STEPS 24
#define QN 9
#define BLK 256

// ---- CDNA5 async global<->LDS path (guarded: falls back cleanly) ----------
#if __has_builtin(__builtin_amdgcn_global_load_async_to_lds_b32) && \
    __has_builtin(__builtin_amdgcn_global_store_async_from_lds_b32) && \
    __has_builtin(__builtin_amdgcn_s_wait_asynccnt)
#define USE_ASYNC_LDS 1
#endif

typedef __attribute__((address_space(1))) int GInt; // global (prints as __device__)
typedef __attribute__((address_space(3))) int LInt; // LDS    (prints as __shared__)

__device__ __forceinline__ GInt* to_global(const void* p) {
  // generic global pointer is bit-identical to AS1 pointer
  return (GInt*)(uintptr_t)p;
}
__device__ __forceinline__ LInt* to_lds(const void* p) {
  // generic LDS aperture pointer: low 32 bits are the LDS offset (ISA 10.2)
  return (LInt*)(uint32_t)(uintptr_t)p;
}

__global__ __launch_bounds__(BLK) void levermore_kernel(
    const float* __restrict__ rho, const float* __restrict__ u,
    const float* __restrict__ T, float* __restrict__ out, int P)
{
  __shared__ float s_in[2 * BLK];      // [0,BLK): rho   [BLK,2BLK): T
  __shared__ float s_out[BLK * QN];    // staged output for coalesced stores

  const int  t     = threadIdx.x;
  const long pBase = (long)blockIdx.x * BLK;
  const long p     = pBase + t;
  const long pc    = (p < P) ? p : (long)(P - 1);

#if defined(USE_ASYNC_LDS)
  // Stage rho and T through the async unit (tracked by ASYNCcnt).
  __builtin_amdgcn_global_load_async_to_lds_b32(to_global(rho + pc),
                                                to_lds(&s_in[t]), 0, 0);
  __builtin_amdgcn_global_load_async_to_lds_b32(to_global(T + pc),
                                                to_lds(&s_in[BLK + t]), 0, 0);
#endif

  // u is interleaved (H,W,2): coalesced 8B vector load.
  const float2 uv = ((const float2*)u)[pc];
  const float ux = uv.x, uy = uv.y;

  float rhov, Tv;
#if defined(USE_ASYNC_LDS)
  __builtin_amdgcn_s_wait_asynccnt(0);   // own-wave async loads complete
  rhov = s_in[t];
  Tv   = s_in[BLK + t];
#else
  rhov = rho[pc];
  Tv   = T[pc];
#endif

  // ---- per-point constants -------------------------------------------------
  const float Ts  = fmaxf(Tv, 1e-6f);
  const float rs  = fmaxf(rhov, 1e-6f);
  const float omt = 1.0f - Ts;
  const float ht  = 0.5f * Ts;
  const float wa  = omt * omt;   // e = ( 0, 0)
  const float wb  = ht * omt;    // axis directions
  const float wc  = ht * ht;     // diagonal directions
  const float uu  = ux * ux + uy * uy;
  const float E   = Ts * CV_F + 0.5f * uu;
  const float Hh  = E + Ts;
  const float tgt0 = 2.0f * E;
  const float tgt1 = 2.0f * ux * Hh;
  const float tgt2 = 2.0f * uy * Hh;

  // ---- Newton iteration on y = (khi, zx, zy) ------------------------------
  float khi = 0.0f, zx = 0.0f, zy = 0.0f;

#pragma unroll 1
  for (int step = 0; step < MAX_STEPS; ++step) {
    const float sdq = zx + zy, ddq = zx - zy;
    const float f0 = wa * __expf(khi);
    const float f1 = wb * __expf(khi + zx);
    const float f2 = wb * __expf(khi + zy);
    const float f3 = wb * __expf(khi - zx);
    const float f4 = wb * __expf(khi - zy);
    const float f5 = wc * __expf(khi + sdq);  // ( 1, 1)
    const float f6 = wc * __expf(khi - ddq);  // (-1, 1)
    const float f7 = wc * __expf(khi - sdq);  // (-1,-1)
    const float f8 = wc * __expf(khi + ddq);  // ( 1,-1)

    // Moments with CSE (E entries are 0/±1 -> pure adds)
    const float a13 = f1 + f3, d13 = f1 - f3;
    const float a24 = f2 + f4, d24 = f2 - f4;
    const float s57 = f5 + f7, d57 = f5 - f7;
    const float s68 = f6 + f8, d68 = f6 - f8;
    const float sdg = s57 + s68;
    const float M0  = f0 + a13 + a24 + sdg;
    const float M1x = d13 + d57 - d68;
    const float M1y = d24 + d57 + d68;
    const float Mxx = a13 + sdg;
    const float Myy = a24 + sdg;
    const float Mxy = s57 - s68;

    const float r0 = M0  - tgt0;
    const float r1 = M1x - tgt1;
    const float r2 = M1y - tgt2;

    // Symmetric 3x3 solve via adjugate (J is SPD)
    const float A00 = Mxx * Myy - Mxy * Mxy;
    const float A01 = M1y * Mxy - M1x * Myy;
    const float A02 = M1x * Mxy - M1y * Mxx;
    const float A11 = M0  * Myy - M1y * M1y;
    const float A12 = M1x * M1y - M0  * Mxy;
    const float A22 = M0  * Mxx - M1x * M1x;
    const float det = M0 * A00 + M1x * A01 + M1y * A02;
    const float inv = 1.0f / det;
    const float dy0 = (A00 * r0 + A01 * r1 + A02 * r2) * inv;
    const float dy1 = (A01 * r0 + A11 * r1 + A12 * r2) * inv;
    const float dy2 = (A02 * r0 + A12 * r1 + A22 * r2) * inv;

    // convergence check uses ||dy|| vs old ||y|| (reference semantics)
    const float ndy = sqrtf(dy0 * dy0 + dy1 * dy1 + dy2 * dy2);
    const float ny  = sqrtf(khi * khi + zx * zx + zy * zy);
    khi -= dy0; zx -= dy1; zy -= dy2;
    if (ndy <= ATOL_F + RTOL_F * ny) break;  // y frozen hereafter == reference mask
  }

  // ---- final distributions, staged to LDS for coalesced streaming ---------
  {
    const float sdq = zx + zy, ddq = zx - zy;
    s_out[t * QN + 0] = rs * wa * __expf(khi);
    s_out[t * QN + 1] = rs * wb * __expf(khi + zx);
    s_out[t * QN + 2] = rs * wb * __expf(khi + zy);
    s_out[t * QN + 3] = rs * wb * __expf(khi - zx);
    s_out[t * QN + 4] = rs * wb * __expf(khi - zy);
    s_out[t * QN + 5] = rs * wc * __expf(khi + sdq);
    s_out[t * QN + 6] = rs * wc * __expf(khi - ddq);
    s_out[t * QN + 7] = rs * wc * __expf(khi - sdq);
    s_out[t * QN + 8] = rs * wc * __expf(khi + ddq);
  }
  __syncthreads();

  const long outBase  = pBase * QN;
  const long outTotal = (long)P * QN;
#pragma unroll
  for (int i = 0; i < QN; ++i) {
    const int  idx = i * BLK + t;       // contiguous across the block
    const long g   = outBase + idx;
    if (g < outTotal) {
#if defined(USE_ASYNC_LDS)
      __builtin_amdgcn_global_store_async_from_lds_b32(to_global(out + g),
                                                       to_lds(&s_out[idx]), 0, 0);
#else
      __builtin_nontemporal_store(s_out[idx], out + g);  // streamed, never re-read
#endif
    }
  }
#if defined(USE_ASYNC_LDS)
  __builtin_amdgcn_s_wait_asynccnt(0);
#endif
}

extern "C" void kernel_launch(void* const* d_in, const int* in_sizes, int n_in,
                              void* d_out, int out_size, void* d_ws, size_t ws_size,
                              hipStream_t stream) {
  (void)n_in; (void)out_size; (void)d_ws; (void)ws_size;
  const float* rho = (const float*)d_in[0];
  const float* u   = (const float*)d_in[1];
  const float* T   = (const float*)d_in[2];
  // d_in[3] is the fixed D2Q9 stencil; its 0/±1 structure is hardcoded.
  const int P = in_sizes[0];
  float* out = (float*)d_out;
  const int blocks = (P + BLK - 1) / BLK;
  hipLaunchKernelGGL(levermore_kernel, dim3(blocks), dim3(BLK), 0, stream,
                     rho, u, T, out, P);
}